// MTCNN_41351945126363
// MI455X (gfx1250) — compile-verified
//
#include <hip/hip_runtime.h>
#include <hip/hip_bf16.h>

#define NBOX   6144
#define NPAD   8192
#define NCB16  384          /* NBOX/16 column blocks of the suppression mask */
#define IOU_THR 0.3f

typedef __attribute__((ext_vector_type(2))) float v2f;
typedef __attribute__((ext_vector_type(8))) float v8f;

// ---------------------------------------------------------------------------
// Kernel 1: bitonic argsort (descending score, stable) fully in LDS + gather.
// key = (~score_bits)<<32 | idx  -> ascending u64 sort == descending score,
// index-ascending tie break; pad entries get key hi = 0xFFFFFFFF, idx >= N.
// ---------------------------------------------------------------------------
__global__ __launch_bounds__(1024) void nms_sort_gather(
    const float* __restrict__ boxes, const float* __restrict__ scores,
    int* __restrict__ order, float* __restrict__ ssort,
    float* __restrict__ sx, float* __restrict__ sy,
    float* __restrict__ sr, float* __restrict__ sb, float* __restrict__ sa)
{
    __shared__ unsigned long long key[NPAD];    // 64 KB of the 320 KB WGP LDS
    const int t = threadIdx.x;
    for (int i = t; i < NPAD; i += 1024) {
        unsigned hi;
        if (i < NBOX) {
            unsigned u = __float_as_uint(scores[i]); // scores >= 0: bits monotone
            hi = ~u;                                  // ascending == score-descending
        } else {
            hi = 0xFFFFFFFFu;
        }
        key[i] = ((unsigned long long)hi << 32) | (unsigned)i;
    }
    __syncthreads();
    for (int k = 2; k <= NPAD; k <<= 1) {
        for (int j = k >> 1; j > 0; j >>= 1) {
            for (int i = t; i < NPAD; i += 1024) {
                int ixj = i ^ j;
                if (ixj > i) {
                    unsigned long long a = key[i], b = key[ixj];
                    bool up = ((i & k) == 0);
                    if (up ? (a > b) : (a < b)) { key[i] = b; key[ixj] = a; }
                }
            }
            __syncthreads();
        }
    }
    for (int i = t; i < NBOX; i += 1024) {
        int oi = (int)(unsigned)key[i];
        order[i] = oi;
        ssort[i] = scores[oi];
        float x = boxes[oi * 4 + 0], y  = boxes[oi * 4 + 1];
        float r = boxes[oi * 4 + 2], bb = boxes[oi * 4 + 3];
        sx[i] = x; sy[i] = y; sr[i] = r; sb[i] = bb;
        sa[i] = (r - x + 1.0f) * (bb - y + 1.0f);
    }
}

// ---------------------------------------------------------------------------
// Kernel 2: suppression bitmask. Block = 8 waves; each wave owns one 16x16
// IoU tile (rows r0..r0+15, cols c0+w*16..+15). Box tiles staged to LDS via
// async-to-LDS; union base (area_i + area_j) computed on the matrix pipe via
// V_WMMA_F32_16X16X4_F32 as an outer product [area_r,1] x [1;area_c].
// ---------------------------------------------------------------------------
__global__ __launch_bounds__(256) void nms_iou_mask(
    const float* __restrict__ sx, const float* __restrict__ sy,
    const float* __restrict__ sr, const float* __restrict__ sb,
    const float* __restrict__ sa, unsigned short* __restrict__ mask)
{
    const int r0 = blockIdx.y * 16;     // 16 rows per block
    const int c0 = blockIdx.x * 128;    // 128 cols per block
    const int t  = threadIdx.x;

    if (c0 + 127 <= r0) {               // strictly lower-triangular: all zero
        if (t < 128) {
            int m = t >> 3, cb = t & 7;
            mask[(size_t)(r0 + m) * NCB16 + (c0 >> 4) + cb] = 0;
        }
        return;
    }

    __shared__ float cx[128], cy[128], cr[128], cbx[128], ca[128];
    __shared__ float rx[16],  ry[16],  rr[16],  rb[16],  ra[16];

#if __has_builtin(__builtin_amdgcn_global_load_async_to_lds_b32)
  #define ASYNC_CP(dst, src)                                                \
    __builtin_amdgcn_global_load_async_to_lds_b32(                          \
        (__attribute__((address_space(1))) int*)(src),                      \
        (__attribute__((address_space(3))) int*)(dst), 0, 0)
    if (t < 128) {
        ASYNC_CP(&cx[t],  &sx[c0 + t]);
        ASYNC_CP(&cy[t],  &sy[c0 + t]);
        ASYNC_CP(&cr[t],  &sr[c0 + t]);
        ASYNC_CP(&cbx[t], &sb[c0 + t]);
        ASYNC_CP(&ca[t],  &sa[c0 + t]);
    } else if (t < 144) {
        int m = t - 128;
        ASYNC_CP(&rx[m], &sx[r0 + m]);
        ASYNC_CP(&ry[m], &sy[r0 + m]);
        ASYNC_CP(&rr[m], &sr[r0 + m]);
        ASYNC_CP(&rb[m], &sb[r0 + m]);
        ASYNC_CP(&ra[m], &sa[r0 + m]);
    }
  #if __has_builtin(__builtin_amdgcn_s_wait_asynccnt)
    __builtin_amdgcn_s_wait_asynccnt(0);
  #else
    asm volatile("s_wait_asynccnt 0" ::: "memory");
  #endif
#else
  #warning "global_load_async_to_lds_b32 builtin missing - direct LDS fill"
    if (t < 128) {
        cx[t] = sx[c0 + t]; cy[t] = sy[c0 + t]; cr[t] = sr[c0 + t];
        cbx[t] = sb[c0 + t]; ca[t] = sa[c0 + t];
    } else if (t < 144) {
        int m = t - 128;
        rx[m] = sx[r0 + m]; ry[m] = sy[r0 + m]; rr[m] = sr[r0 + m];
        rb[m] = sb[r0 + m]; ra[m] = sa[r0 + m];
    }
#endif
    __syncthreads();

    const int lane = t & 31;
    const int w    = t >> 5;                  // wave id 0..7 -> 16-col slice
    const int hi   = lane >> 4;               // half-wave
    const int nl   = (w << 4) | (lane & 15);  // local col 0..127

    // A(16x4): K0 = row areas, K1 = 1 (lanes 0-15 hold K0/K1; 16-31 K2/K3 = 0)
    // B(4x16): K0 = 1, K1 = col areas  -> D[m][n] = area_row[m] + area_col[n]
    v2f a, b;
    a.x = (hi == 0) ? ra[lane & 15] : 0.0f;
    a.y = (hi == 0) ? 1.0f          : 0.0f;
    b.x = (hi == 0) ? 1.0f          : 0.0f;
    b.y = (hi == 0) ? ca[nl]        : 0.0f;
    v8f acc = {};
#if __has_builtin(__builtin_amdgcn_wmma_f32_16x16x4_f32)
    acc = __builtin_amdgcn_wmma_f32_16x16x4_f32(false, a, false, b,
                                                (short)0, acc, false, false);
#else
  #warning "wmma_f32_16x16x4_f32 builtin missing - VALU fallback"
    #pragma unroll
    for (int v = 0; v < 8; ++v) acc[v] = ra[v + (hi << 3)] + ca[nl];
#endif

    const float cxv = cx[nl], cyv = cy[nl], crv = cr[nl], cbv = cbx[nl];
    const int   gc   = c0 + nl;
    const int   cb16 = (c0 >> 4) + w;

    #pragma unroll
    for (int v = 0; v < 8; ++v) {
        int   m   = v + (hi << 3);            // C/D layout: rows v / v+8
        float xm  = fmaxf(rx[m], cxv);
        float ym  = fmaxf(ry[m], cyv);
        float xM  = fminf(rr[m], crv);
        float yM  = fminf(rb[m], cbv);
        float iw  = xM - xm + 1.0f;
        float ih  = yM - ym + 1.0f;
        float cross = ((iw > 0.0f) && (ih > 0.0f)) ? iw * ih : 0.0f;
        float uni   = acc[v] - cross + 1e-6f;
        bool  dec   = ((cross / uni) > IOU_THR) && (gc > (r0 + m));
        unsigned bal = __builtin_amdgcn_ballot_w32(dec);
        if (lane == v)
            mask[(size_t)(r0 + v)     * NCB16 + cb16] = (unsigned short)(bal & 0xFFFFu);
        if (lane == v + 8)
            mask[(size_t)(r0 + v + 8) * NCB16 + cb16] = (unsigned short)(bal >> 16);
    }
}

// ---------------------------------------------------------------------------
// Kernel 3: sequential greedy scan. One block, 384 threads; each thread owns
// one 16-bit "removed" word in LDS. Then emit both outputs.
// ---------------------------------------------------------------------------
__global__ __launch_bounds__(384) void nms_scan(
    const unsigned short* __restrict__ mask, const int* __restrict__ order,
    const float* __restrict__ ssort, float* __restrict__ out)
{
    __shared__ unsigned rem[NCB16];
    __shared__ int flag;
    const int t = threadIdx.x;          // 0..383
    rem[t] = 0;
    __syncthreads();
    for (int i = 0; i < NBOX; ++i) {
        if (t == 0) flag = ((rem[i >> 4] >> (i & 15)) & 1u) ? 0 : 1;
        __syncthreads();
        if (flag) rem[t] |= (unsigned)mask[(size_t)i * NCB16 + t];
        __syncthreads();
    }
    for (int i = t; i < NBOX; i += 384) {
        bool keep = (((rem[i >> 4] >> (i & 15)) & 1u) == 0);
        out[i]                 = keep ? ssort[i] : 0.0f;   // kept_scores (sorted)
        out[NBOX + order[i]]   = keep ? 1.0f : 0.0f;       // keep mask (orig order)
    }
}

// ---------------------------------------------------------------------------
extern "C" void kernel_launch(void* const* d_in, const int* in_sizes, int n_in,
                              void* d_out, int out_size, void* d_ws, size_t ws_size,
                              hipStream_t stream) {
    const float* boxes  = (const float*)d_in[0];   // (N,4) x,y,r,b
    const float* scores = (const float*)d_in[1];   // (N,)
    float* out = (float*)d_out;                    // [kept_scores | keep_orig]

    char* w = (char*)d_ws;
    int*   order = (int*)w;            w += (size_t)NBOX * sizeof(int);
    float* ssort = (float*)w;          w += (size_t)NBOX * sizeof(float);
    float* sx    = (float*)w;          w += (size_t)NBOX * sizeof(float);
    float* sy    = (float*)w;          w += (size_t)NBOX * sizeof(float);
    float* sr    = (float*)w;          w += (size_t)NBOX * sizeof(float);
    float* sb    = (float*)w;          w += (size_t)NBOX * sizeof(float);
    float* sa    = (float*)w;          w += (size_t)NBOX * sizeof(float);
    unsigned short* mask = (unsigned short*)w;     // NBOX * 384 halfwords (~4.7MB)

    nms_sort_gather<<<1, 1024, 0, stream>>>(boxes, scores, order, ssort,
                                            sx, sy, sr, sb, sa);
    nms_iou_mask<<<dim3(NBOX / 128, NBOX / 16), 256, 0, stream>>>(
        sx, sy, sr, sb, sa, mask);
    nms_scan<<<1, 384, 0, stream>>>(mask, order, ssort, out);
}